// Soma_83176336654847
// MI455X (gfx1250) — compile-verified
//
#include <hip/hip_runtime.h>

// LIF soma forward: o[t] = heaviside(u[t] - U_thr), hard reset.
// Memory-bound streaming scan: 256 MB total traffic, ~11 us at 23.3 TB/s.
// gfx1250 paths used: wave32 + B128 vector loads/stores, non-temporal cache
// hints (one-touch data vs 192MB L2), and global_prefetch_b8 for the
// 2MB-strided timestep-(t+2) line (address independent of the recurrence).

typedef float v4f __attribute__((ext_vector_type(4)));

#define U_THRESHOLD (-0.055f)
#define U_REST      (-0.07f)
#define T_STEPS     64

__global__ __launch_bounds__(256) void lif_soma_fwd_v4(
    const v4f* __restrict__ x4,      // [T, BN/4]
    const float* __restrict__ tau_m, // [1]
    v4f* __restrict__ o4,            // [T, BN/4]
    int bn4)                         // BN/4
{
    int idx = blockIdx.x * 256 + threadIdx.x;
    if (idx >= bn4) return;

    const float inv_tau = 1.0f / tau_m[0];   // uniform; SMEM load + scalar rcp

    v4f h;
    h.x = U_REST; h.y = U_REST; h.z = U_REST; h.w = U_REST;

    size_t off = (size_t)idx;
    const size_t stride = (size_t)bn4;       // one timestep = BN floats

#pragma unroll 4
    for (int t = 0; t < T_STEPS; ++t) {
        // Prefetch the timestep-(t+2) line: two timesteps away, but the address does
        // not depend on the serial recurrence -> global_prefetch_b8.
        if (t + 2 < T_STEPS)
            __builtin_prefetch((const void*)(x4 + off + 2 * stride), 0, 0);

        // One-touch streaming data: non-temporal B128 load.
        v4f xv = __builtin_nontemporal_load(x4 + off);

        // u = h + (1/tau) * (U_REST - h + x)
        v4f u = h + inv_tau * (U_REST - h + xv);

        // o = heaviside(u - U_thr); h_next = spike ? U_REST : u  (hard reset)
        v4f o;
        o.x = (u.x >= U_THRESHOLD) ? 1.0f : 0.0f;
        o.y = (u.y >= U_THRESHOLD) ? 1.0f : 0.0f;
        o.z = (u.z >= U_THRESHOLD) ? 1.0f : 0.0f;
        o.w = (u.w >= U_THRESHOLD) ? 1.0f : 0.0f;

        h.x = (u.x >= U_THRESHOLD) ? U_REST : u.x;
        h.y = (u.y >= U_THRESHOLD) ? U_REST : u.y;
        h.z = (u.z >= U_THRESHOLD) ? U_REST : u.z;
        h.w = (u.w >= U_THRESHOLD) ? U_REST : u.w;

        // One-touch output: non-temporal B128 store.
        __builtin_nontemporal_store(o, o4 + off);

        off += stride;
    }
}

// Scalar tail for BN not divisible by 4 (not hit for the reference shape).
__global__ __launch_bounds__(64) void lif_soma_fwd_tail(
    const float* __restrict__ x,
    const float* __restrict__ tau_m,
    float* __restrict__ out,
    int bn, int start)
{
    int i = start + blockIdx.x * 64 + threadIdx.x;
    if (i >= bn) return;

    const float inv_tau = 1.0f / tau_m[0];
    float h = U_REST;
    size_t off = (size_t)i;
    for (int t = 0; t < T_STEPS; ++t) {
        float xv = __builtin_nontemporal_load(x + off);
        float u  = h + inv_tau * (U_REST - h + xv);
        float o  = (u >= U_THRESHOLD) ? 1.0f : 0.0f;
        __builtin_nontemporal_store(o, out + off);
        h = (u >= U_THRESHOLD) ? U_REST : u;
        off += (size_t)bn;
    }
}

extern "C" void kernel_launch(void* const* d_in, const int* in_sizes, int n_in,
                              void* d_out, int out_size, void* d_ws, size_t ws_size,
                              hipStream_t stream) {
    const float* x     = (const float*)d_in[0];   // [T, B, N] f32
    const float* tau_m = (const float*)d_in[1];   // [1] f32
    float* out         = (float*)d_out;           // [T, B, N] f32

    const long long total = (long long)in_sizes[0];
    const int bn  = (int)(total / T_STEPS);       // B*N neurons
    const int bn4 = bn / 4;
    const int rem = bn - bn4 * 4;

    if (bn4 > 0) {
        const int threads = 256;
        const int blocks  = (bn4 + threads - 1) / threads;
        lif_soma_fwd_v4<<<blocks, threads, 0, stream>>>(
            (const v4f*)x, tau_m, (v4f*)out, bn4);
    }
    if (rem > 0) {
        lif_soma_fwd_tail<<<1, 64, 0, stream>>>(x, tau_m, out, bn, bn4 * 4);
    }
}